// AttentionPooling_26061861552522
// MI455X (gfx1250) — compile-verified
//
#include <hip/hip_runtime.h>
#include <hip/hip_bf16.h>
#include <math.h>

#define BATCH 256
#define SLEN  2500
#define FD    64
#define NH    8
#define HD    8
#define INV_SQRT8 0.35355339059327373f

typedef __attribute__((ext_vector_type(2))) float v2f;
typedef __attribute__((ext_vector_type(8))) float v8f;

// ---------------------------------------------------------------------------
// Kernel P: per-position score coefficients (batch-independent, done once).
//   scores[h,t] = a^2*C2[h,t] + a*C1_s[h,t] + C0_s[h,t]   (1/sqrt(8) folded in)
// Layout: c01[s*128 + h*16 + t]     = C1_s[h,t]
//         c01[s*128 + h*16 + 8 + t] = C0_s[h,t]
// c2[h*8+t] = C2[h,t] (uniform, read via scalar loads in the attention kernel)
// ---------------------------------------------------------------------------
__global__ __launch_bounds__(256) void precompute_kernel(
    const float* __restrict__ wq, const float* __restrict__ bq,
    const float* __restrict__ wk, const float* __restrict__ bk,
    const float* __restrict__ pos_enc,
    float* __restrict__ c01, float* __restrict__ c2)
{
    const int tid = threadIdx.x;
    if (blockIdx.x == 0 && tid < NH * NH) {
        const int h = tid >> 3, t = tid & 7;
        float d = 0.0f;
#pragma unroll
        for (int i = 0; i < HD; ++i) d = fmaf(wq[h * HD + i], wk[t * HD + i], d);
        c2[tid] = d * INV_SQRT8;
    }

    const int s = blockIdx.x * blockDim.x + tid;
    if (s >= SLEN) return;

    const float* __restrict__ pe = pos_enc + s * FD;
    float qb[FD], kb[FD];
#pragma unroll
    for (int i = 0; i < FD; ++i) {
        qb[i] = bq[i] + pe[i];
        kb[i] = bk[i] + pe[i];
    }

    float* __restrict__ row = c01 + s * 128;
#pragma unroll
    for (int h = 0; h < NH; ++h) {
#pragma unroll
        for (int t = 0; t < NH; ++t) {
            float c0 = 0.0f, cq = 0.0f, ck = 0.0f;
#pragma unroll
            for (int d = 0; d < HD; ++d) {
                c0 = fmaf(qb[h * HD + d], kb[t * HD + d], c0);
                cq = fmaf(wq[h * HD + d], kb[t * HD + d], cq);
                ck = fmaf(wk[t * HD + d], qb[h * HD + d], ck);
            }
            row[h * 16 + t]     = (cq + ck) * INV_SQRT8;  // C1
            row[h * 16 + 8 + t] = c0 * INV_SQRT8;         // C0
        }
    }
}

// ---------------------------------------------------------------------------
// Kernel A: per-position 8-head micro-attention + mean-pool over S.
// One block per batch b; lane-serial (wave32 f32 VALU), scores rebuilt from the
// quadratic-in-a coefficient tables (L2-resident).
// ---------------------------------------------------------------------------
__global__ __launch_bounds__(512) void attn_pool_kernel(
    const float* __restrict__ sim,      // [B, S]
    const float* __restrict__ wv, const float* __restrict__ bv,
    const float* __restrict__ c01,      // [S, 128]
    const float* __restrict__ c2,       // [64] uniform
    float* __restrict__ pooled)         // [B, FD]
{
    __shared__ float acc[FD];
    const int b   = blockIdx.x;
    const int tid = threadIdx.x;

    for (int i = tid; i < FD; i += blockDim.x) acc[i] = 0.0f;
    __syncthreads();

    float att[FD];
#pragma unroll
    for (int i = 0; i < FD; ++i) att[i] = 0.0f;

    for (int s = tid; s < SLEN; s += blockDim.x) {
        const float a  = sim[b * SLEN + s];
        const float a2 = a * a;

        if (s + (int)blockDim.x < SLEN)
            __builtin_prefetch(c01 + (s + blockDim.x) * 128, 0, 1);

        float vv[FD];
#pragma unroll
        for (int i = 0; i < FD; ++i) vv[i] = fmaf(a, wv[i], bv[i]);

        const float4* __restrict__ row4 =
            reinterpret_cast<const float4*>(c01 + s * 128);

#pragma unroll
        for (int h = 0; h < NH; ++h) {
            const float4 x0 = row4[h * 4 + 0];   // C1[0..3]
            const float4 x1 = row4[h * 4 + 1];   // C1[4..7]
            const float4 x2 = row4[h * 4 + 2];   // C0[0..3]
            const float4 x3 = row4[h * 4 + 3];   // C0[4..7]
            float c1[NH] = {x0.x, x0.y, x0.z, x0.w, x1.x, x1.y, x1.z, x1.w};
            float c0[NH] = {x2.x, x2.y, x2.z, x2.w, x3.x, x3.y, x3.z, x3.w};

            float sc[NH];
            float m = -INFINITY;
#pragma unroll
            for (int t = 0; t < NH; ++t) {
                const float v = fmaf(a2, c2[h * NH + t], fmaf(a, c1[t], c0[t]));
                sc[t] = v;
                m = fmaxf(m, v);
            }
            float sum = 0.0f;
#pragma unroll
            for (int t = 0; t < NH; ++t) {
                sc[t] = __expf(sc[t] - m);
                sum += sc[t];
            }
            const float inv = __builtin_amdgcn_rcpf(sum);
#pragma unroll
            for (int d = 0; d < HD; ++d) {
                float o = 0.0f;
#pragma unroll
                for (int t = 0; t < NH; ++t)
                    o = fmaf(sc[t], vv[t * HD + d], o);
                att[h * HD + d] = fmaf(o, inv, att[h * HD + d]);
            }
        }
    }

#pragma unroll
    for (int i = 0; i < FD; ++i) atomicAdd(&acc[i], att[i]);
    __syncthreads();

    const float inv_s = 1.0f / (float)SLEN;
    for (int i = tid; i < FD; i += blockDim.x)
        pooled[b * FD + i] = acc[i] * inv_s;
}

// ---------------------------------------------------------------------------
// Kernel 2: out_pre = pooled(256x64) @ Wo^T(64x64) + bo, via
// V_WMMA_F32_16X16X4_F32. One wave per 16x16 C tile; K=64 in 16 steps of 4.
// Layouts per CDNA5 ISA 7.12.2:
//   A: lane L, reg j -> A[M=L%16][K = 2*(L/16)+j]
//   B: lane L, reg j -> B[K = 2*(L/16)+j][N=L%16]
//   C: lane L, reg r -> C[M = r + 8*(L/16)][N=L%16]
// ---------------------------------------------------------------------------
__global__ __launch_bounds__(32) void proj_wmma_kernel(
    const float* __restrict__ pooled,   // [256, 64]
    const float* __restrict__ Wo,       // [64, 64]  (out = pooled @ Wo^T)
    const float* __restrict__ bo,       // [64]
    float* __restrict__ out_pre)        // [256, 64]
{
    const int lane = threadIdx.x;       // 0..31
    const int mt   = blockIdx.x * 16;   // row tile base (M)
    const int nt   = blockIdx.y * 16;   // col tile base (N)
    const int l16  = lane & 15;
    const int half = lane >> 4;         // 0 or 1

    v8f c = {};
#pragma unroll
    for (int ks = 0; ks < FD; ks += 4) {
        const int k0 = ks + half * 2;
        v2f amat, bmat;
        amat[0] = pooled[(mt + l16) * FD + k0 + 0];
        amat[1] = pooled[(mt + l16) * FD + k0 + 1];
        // B[k][n] = Wo^T[k][n] = Wo[n][k]
        bmat[0] = Wo[(nt + l16) * FD + k0 + 0];
        bmat[1] = Wo[(nt + l16) * FD + k0 + 1];
        c = __builtin_amdgcn_wmma_f32_16x16x4_f32(
                /*neg_a=*/false, amat, /*neg_b=*/false, bmat,
                /*c_mod=*/(short)0, c, /*reuse_a=*/false, /*reuse_b=*/false);
    }

    const int col = nt + l16;
    const float bias = bo[col];
#pragma unroll
    for (int r = 0; r < 8; ++r) {
        const int row = mt + r + 8 * half;
        out_pre[row * FD + col] = c[r] + bias;
    }
}

// ---------------------------------------------------------------------------
// Kernel 3: LayerNorm over the 64-feature dim. One block (64 threads) per row.
// ---------------------------------------------------------------------------
__global__ __launch_bounds__(64) void layernorm_kernel(
    const float* __restrict__ x,        // [256, 64]
    const float* __restrict__ gamma,
    const float* __restrict__ beta,
    float* __restrict__ out)            // [256, 64]
{
    __shared__ float r1[FD];
    __shared__ float r2[FD];
    const int b = blockIdx.x;
    const int t = threadIdx.x;

    const float v = x[b * FD + t];
    r1[t] = v;
    r2[t] = v * v;
    __syncthreads();

#pragma unroll
    for (int off = 32; off > 0; off >>= 1) {
        if (t < off) {
            r1[t] += r1[t + off];
            r2[t] += r2[t + off];
        }
        __syncthreads();
    }

    const float mu  = r1[0] * (1.0f / FD);
    const float var = r2[0] * (1.0f / FD) - mu * mu;
    const float inv = rsqrtf(var + 1e-5f);
    out[b * FD + t] = (v - mu) * inv * gamma[t] + beta[t];
}

// ---------------------------------------------------------------------------
extern "C" void kernel_launch(void* const* d_in, const int* in_sizes, int n_in,
                              void* d_out, int out_size, void* d_ws, size_t ws_size,
                              hipStream_t stream) {
    const float* sim     = (const float*)d_in[0];
    const float* wq      = (const float*)d_in[1];
    const float* bq      = (const float*)d_in[2];
    const float* wk      = (const float*)d_in[3];
    const float* bk      = (const float*)d_in[4];
    const float* wv      = (const float*)d_in[5];
    const float* bv      = (const float*)d_in[6];
    const float* pos_enc = (const float*)d_in[7];
    const float* Wo      = (const float*)d_in[8];
    const float* bo      = (const float*)d_in[9];
    const float* gamma   = (const float*)d_in[10];
    const float* beta    = (const float*)d_in[11];
    float* out           = (float*)d_out;

    // Workspace layout (floats): c01 [S*128] | c2 [64] | pooled [B*64] | pre [B*64]
    float* c01     = (float*)d_ws;
    float* c2      = c01 + SLEN * 128;
    float* pooled  = c2 + FD;
    float* out_pre = pooled + BATCH * FD;

    precompute_kernel<<<(SLEN + 255) / 256, 256, 0, stream>>>(
        wq, bq, wk, bk, pos_enc, c01, c2);

    attn_pool_kernel<<<BATCH, 512, 0, stream>>>(sim, wv, bv, c01, c2, pooled);

    dim3 gtiles(BATCH / 16, FD / 16);              // 16 x 4 waves
    proj_wmma_kernel<<<gtiles, 32, 0, stream>>>(pooled, Wo, bo, out_pre);

    layernorm_kernel<<<BATCH, 64, 0, stream>>>(out_pre, gamma, beta, out);
}